// HAGERecConv_59760174957271
// MI455X (gfx1250) — compile-verified
//
#include <hip/hip_runtime.h>
#include <hip/hip_bf16.h>

// ---------------------------------------------------------------------------
// HAGERecConv on MI455X (gfx1250): wave32 edge pipeline + f32 WMMA projections
// ---------------------------------------------------------------------------
// Sizes (from reference): N=50000, E=1.6M, H=2, F=32, IN=OUT=32.
// Memory-bound on r_feat stream (410MB). WMMA used for the 3 dense GEMMs.

#define H_HEADS 2
#define F_FEAT  32
#define IN_F    32
#define NEG_SLOPE 0.2f

typedef __attribute__((ext_vector_type(2))) float v2f;
typedef __attribute__((ext_vector_type(8))) float v8f;

// ------------------------------ helpers ------------------------------------

__device__ __forceinline__ float waveReduceSum(float v) {
#pragma unroll
    for (int off = 16; off > 0; off >>= 1) v += __shfl_xor(v, off, 32);
    return v;
}

__device__ __forceinline__ float waveReduceMax(float v) {
#pragma unroll
    for (int off = 16; off > 0; off >>= 1) v = fmaxf(v, __shfl_xor(v, off, 32));
    return v;
}

// float atomic max via sign-split int atomics (works for any sign, init = -inf)
__device__ __forceinline__ void atomicMaxF(float* addr, float val) {
    if (val >= 0.0f) atomicMax((int*)addr, __float_as_int(val));
    else             atomicMin((unsigned int*)addr, __float_as_uint(val));
}

// ------------------------ init workspace buffers ---------------------------

__global__ void init_buffers_kernel(float* __restrict__ m_buf,
                                    float* __restrict__ denom,
                                    float* __restrict__ g_sum,
                                    int n_mh, int n_gf) {
    int i = blockIdx.x * blockDim.x + threadIdx.x;
    if (i < n_mh) {
        m_buf[i] = -__builtin_huge_valf();  // segment_max identity
        denom[i] = 0.0f;
    }
    if (i < n_gf) g_sum[i] = 0.0f;
}

// ---------------- GEMM 1: feat @ [W2s | W2d] + bias  (WMMA f32) ------------
// One wave per 16x16 output tile. 8 column tiles: 0-3 -> feat_src (W2s),
// 4-7 -> feat_dst (W2d). K = 32 done as 8 x V_WMMA_F32_16X16X4_F32.
// A frag (16x4 f32, 2 VGPRs): lane m in [0,16): (M=m, K=k0+{0,1});
//                             lane m in [16,32): (M=m-16, K=k0+{2,3}).
// B frag (4x16): VGPRv holds row K=k0+2*half+v, col N = lane&15.
// C/D (16x16, 8 VGPRs): elem (M = v + 8*(lane>>4), N = lane&15).

__global__ void proj_gemm_kernel(const float* __restrict__ feat,
                                 const float* __restrict__ W2s,
                                 const float* __restrict__ b2s,
                                 const float* __restrict__ W2d,
                                 const float* __restrict__ b2d,
                                 float* __restrict__ feat_src,
                                 float* __restrict__ feat_dst,
                                 int n_nodes) {
    const int gtid = blockIdx.x * blockDim.x + threadIdx.x;
    const int wave = gtid >> 5;                 // global wave id (uniform)
    const int lane = gtid & 31;
    const int nRowTiles = n_nodes >> 4;         // N multiple of 16
    if (wave >= nRowTiles * 8) return;          // uniform per wave: EXEC all-1

    const int rowTile = wave >> 3;
    const int ct      = wave & 7;               // column tile 0..7
    const float* __restrict__ W    = (ct < 4) ? W2s : W2d;
    const float* __restrict__ bias = (ct < 4) ? b2s : b2d;
    float* __restrict__ outp       = (ct < 4) ? feat_src : feat_dst;
    const int c0   = (ct & 3) * 16;             // column offset within [0,64)
    const int row0 = rowTile * 16;

    const int half = lane >> 4;                 // 0 or 1
    const int mr   = lane & 15;                 // A row / B,C col index

    const float* __restrict__ Arow = feat + (size_t)(row0 + mr) * IN_F;

    v8f acc = {};
#pragma unroll
    for (int k0 = 0; k0 < IN_F; k0 += 4) {
        const int ka = k0 + half * 2;
        v2f a, b;
        a.x = Arow[ka];
        a.y = Arow[ka + 1];
        b.x = W[(size_t)ka       * (H_HEADS * F_FEAT) + c0 + mr];
        b.y = W[(size_t)(ka + 1) * (H_HEADS * F_FEAT) + c0 + mr];
        acc = __builtin_amdgcn_wmma_f32_16x16x4_f32(
                  false, a, false, b, (short)0, acc, false, false);
    }

    const float bv = bias[c0 + mr];
#pragma unroll
    for (int v = 0; v < 8; ++v) {
        const int row = row0 + v + 8 * half;
        outp[(size_t)row * (H_HEADS * F_FEAT) + c0 + mr] = acc[v] + bv;
    }
}

// ---------------- E1: per-edge attention logit + segment max ---------------
// One wave per edge; F=32 lanes. e = sum_F(leakyrelu(en+ev)*attn) per head.

__global__ void edge_logit_kernel(const float* __restrict__ feat_src,
                                  const float* __restrict__ feat_dst,
                                  const int* __restrict__ src,
                                  const int* __restrict__ dst,
                                  const float* __restrict__ attn,
                                  float* __restrict__ e_buf,
                                  float* __restrict__ m_buf,
                                  int n_edges) {
    const int gtid = blockIdx.x * blockDim.x + threadIdx.x;
    const int eid  = gtid >> 5;
    const int lane = gtid & 31;
    if (eid >= n_edges) return;

    const int s = src[eid];
    const int d = dst[eid];
#pragma unroll
    for (int h = 0; h < H_HEADS; ++h) {
        float en = feat_src[(size_t)s * (H_HEADS * F_FEAT) + h * F_FEAT + lane];
        float ev = feat_dst[(size_t)d * (H_HEADS * F_FEAT) + h * F_FEAT + lane];
        float x  = en + ev;
        x = (x > 0.0f) ? x : NEG_SLOPE * x;
        x *= attn[h * F_FEAT + lane];
        float e = waveReduceSum(x);
        if (lane == 0) {
            e_buf[(size_t)eid * H_HEADS + h] = e;
            atomicMaxF(&m_buf[(size_t)d * H_HEADS + h], e);
        }
    }
}

// ---------------- E2: ee = exp(e - m[dst]); denom = segment_sum ------------

__global__ void edge_exp_kernel(const int* __restrict__ dst,
                                float* __restrict__ e_buf,   // in: e, out: ee
                                const float* __restrict__ m_buf,
                                float* __restrict__ denom,
                                int n_eh) {
    const int i = blockIdx.x * blockDim.x + threadIdx.x;
    if (i >= n_eh) return;
    const int eid = i >> 1;
    const int h   = i & 1;
    const int d   = dst[eid];
    const float ee = __expf(e_buf[i] - m_buf[(size_t)d * H_HEADS + h]);
    e_buf[i] = ee;
    atomicAdd(&denom[(size_t)d * H_HEADS + h], ee);
}

// ---------------- E3: message softmax + scatter-add (the 410MB stream) -----
// Wave per edge; per head: t = en * r_feat * alpha, softmax over F (lanes),
// fold heads in-register, single atomicAdd per lane into g_sum[dst].

__global__ void edge_message_kernel(const float* __restrict__ feat_src,
                                    const float* __restrict__ r_feat,
                                    const int* __restrict__ src,
                                    const int* __restrict__ dst,
                                    const float* __restrict__ ee_buf,
                                    const float* __restrict__ denom,
                                    float* __restrict__ g_sum,
                                    int n_edges) {
    const int gtid = blockIdx.x * blockDim.x + threadIdx.x;
    const int eid  = gtid >> 5;
    const int lane = gtid & 31;
    if (eid >= n_edges) return;

    const int s = src[eid];
    const int d = dst[eid];

    float acc = 0.0f;
#pragma unroll
    for (int h = 0; h < H_HEADS; ++h) {
        const float alpha = ee_buf[(size_t)eid * H_HEADS + h] /
                            fmaxf(denom[(size_t)d * H_HEADS + h], 1e-9f);
        const float en = feat_src[(size_t)s * (H_HEADS * F_FEAT) + h * F_FEAT + lane];
        const float r  = r_feat[((size_t)eid * H_HEADS + h) * F_FEAT + lane];
        const float t  = en * r * alpha;
        // softmax over feature dim (exactly one wave32)
        const float mx = waveReduceMax(t);
        const float ex = __expf(t - mx);
        const float sm = waveReduceSum(ex);
        acc += ex / sm;                         // head-sum folded in-register
    }
    atomicAdd(&g_sum[(size_t)d * F_FEAT + lane], acc);
}

// ---------------- GEMM 2: (feat + g_sum) @ W1 + b1  (WMMA f32) -------------

__global__ void out_gemm_kernel(const float* __restrict__ feat,
                                const float* __restrict__ g_sum,
                                const float* __restrict__ W1,
                                const float* __restrict__ b1,
                                float* __restrict__ out,
                                int n_nodes) {
    const int gtid = blockIdx.x * blockDim.x + threadIdx.x;
    const int wave = gtid >> 5;
    const int lane = gtid & 31;
    const int nRowTiles = n_nodes >> 4;
    if (wave >= nRowTiles * 2) return;          // uniform per wave

    const int rowTile = wave >> 1;
    const int ct      = wave & 1;
    const int c0   = ct * 16;
    const int row0 = rowTile * 16;

    const int half = lane >> 4;
    const int mr   = lane & 15;

    const float* __restrict__ Arow = feat  + (size_t)(row0 + mr) * IN_F;
    const float* __restrict__ Grow = g_sum + (size_t)(row0 + mr) * F_FEAT;

    v8f acc = {};
#pragma unroll
    for (int k0 = 0; k0 < IN_F; k0 += 4) {
        const int ka = k0 + half * 2;
        v2f a, b;
        a.x = Arow[ka]     + Grow[ka];
        a.y = Arow[ka + 1] + Grow[ka + 1];
        b.x = W1[(size_t)ka       * 32 + c0 + mr];
        b.y = W1[(size_t)(ka + 1) * 32 + c0 + mr];
        acc = __builtin_amdgcn_wmma_f32_16x16x4_f32(
                  false, a, false, b, (short)0, acc, false, false);
    }

    const float bv = b1[c0 + mr];
#pragma unroll
    for (int v = 0; v < 8; ++v) {
        const int row = row0 + v + 8 * half;
        out[(size_t)row * 32 + c0 + mr] = acc[v] + bv;
    }
}

// ------------------------------ launcher -----------------------------------

extern "C" void kernel_launch(void* const* d_in, const int* in_sizes, int n_in,
                              void* d_out, int out_size, void* d_ws, size_t ws_size,
                              hipStream_t stream) {
    const float* feat   = (const float*)d_in[0];   // [N,32]
    const float* r_feat = (const float*)d_in[1];   // [E,2,32]
    const int*   src    = (const int*)  d_in[2];   // [E]
    const int*   dst    = (const int*)  d_in[3];   // [E]
    const float* W1     = (const float*)d_in[4];   // [32,32]
    const float* b1     = (const float*)d_in[5];   // [32]
    const float* W2s    = (const float*)d_in[6];   // [32,64]
    const float* b2s    = (const float*)d_in[7];   // [64]
    const float* W2d    = (const float*)d_in[8];   // [32,64]
    const float* b2d    = (const float*)d_in[9];   // [64]
    const float* attn   = (const float*)d_in[10];  // [1,2,32]

    const int N = in_sizes[0] / IN_F;
    const int E = in_sizes[2];

    // workspace layout (floats)
    float* ws       = (float*)d_ws;
    float* feat_src = ws;                                   // N*64
    float* feat_dst = feat_src + (size_t)N * H_HEADS * F_FEAT;
    float* e_buf    = feat_dst + (size_t)N * H_HEADS * F_FEAT;  // E*2
    float* m_buf    = e_buf    + (size_t)E * H_HEADS;           // N*2
    float* denom    = m_buf    + (size_t)N * H_HEADS;           // N*2
    float* g_sum    = denom    + (size_t)N * H_HEADS;           // N*32

    const int BLK = 256;  // 8 waves per block on wave32

    // 0) init accumulators (m=-inf, denom=0, g_sum=0)
    {
        int n_mh = N * H_HEADS, n_gf = N * F_FEAT;
        int n = (n_mh > n_gf) ? n_mh : n_gf;
        init_buffers_kernel<<<(n + BLK - 1) / BLK, BLK, 0, stream>>>(
            m_buf, denom, g_sum, n_mh, n_gf);
    }
    // 1) node projections via WMMA (3125 row tiles x 8 col tiles)
    {
        int waves = (N >> 4) * 8;
        proj_gemm_kernel<<<(waves * 32 + BLK - 1) / BLK, BLK, 0, stream>>>(
            feat, W2s, b2s, W2d, b2d, feat_src, feat_dst, N);
    }
    // 2) edge logits + segment max (wave per edge)
    {
        long long thr = (long long)E * 32;
        edge_logit_kernel<<<(unsigned)((thr + BLK - 1) / BLK), BLK, 0, stream>>>(
            feat_src, feat_dst, src, dst, attn, e_buf, m_buf, E);
    }
    // 3) exp + segment sum (thread per edge*head)
    {
        int n_eh = E * H_HEADS;
        edge_exp_kernel<<<(n_eh + BLK - 1) / BLK, BLK, 0, stream>>>(
            dst, e_buf, m_buf, denom, n_eh);
    }
    // 4) message softmax + scatter-add (wave per edge, streams r_feat)
    {
        long long thr = (long long)E * 32;
        edge_message_kernel<<<(unsigned)((thr + BLK - 1) / BLK), BLK, 0, stream>>>(
            feat_src, r_feat, src, dst, e_buf, denom, g_sum, E);
    }
    // 5) output GEMM via WMMA
    {
        int waves = (N >> 4) * 2;
        out_gemm_kernel<<<(waves * 32 + BLK - 1) / BLK, BLK, 0, stream>>>(
            feat, g_sum, W1, b1, (float*)d_out, N);
    }
}